// QuantumHead_5738076308065
// MI455X (gfx1250) — compile-verified
//
#include <hip/hip_runtime.h>
#include <hip/hip_bf16.h>

typedef float v2f __attribute__((ext_vector_type(2)));
typedef float v8f __attribute__((ext_vector_type(8)));

#define NQ 4
#define DIN 512
#define ROWS_PER_BLOCK 128   // 8 waves * 16-row WMMA tile
#define KCHUNKS (DIN / 4)    // 128 chunks of K=4 for V_WMMA_F32_16X16X4_F32

// Apply a 2x2 complex gate U=[[u00,u01],[u10,u11]] on the qubit selected by `bit`.
// Fully unrolled at call sites so the 16-amplitude state stays in VGPRs.
__device__ __forceinline__ void apply_gate(float* sr, float* si, int bit,
    float u00r, float u00i, float u01r, float u01i,
    float u10r, float u10i, float u11r, float u11i) {
#pragma unroll
  for (int i0 = 0; i0 < 16; ++i0) {
    if (i0 & bit) continue;
    int i1 = i0 | bit;
    float ar = sr[i0], ai = si[i0], br = sr[i1], bi = si[i1];
    sr[i0] = u00r * ar - u00i * ai + u01r * br - u01i * bi;
    si[i0] = u00r * ai + u00i * ar + u01r * bi + u01i * br;
    sr[i1] = u10r * ar - u10i * ai + u11r * br - u11i * bi;
    si[i1] = u10r * ai + u10i * ar + u11r * bi + u11i * br;
  }
}

__device__ __forceinline__ void cnot_gate(float* sr, float* si, int cb, int tb) {
#pragma unroll
  for (int i = 0; i < 16; ++i) {
    if ((i & cb) && !(i & tb)) {
      int j = i | tb;
      float tr = sr[i], ti = si[i];
      sr[i] = sr[j]; si[i] = si[j];
      sr[j] = tr;    si[j] = ti;
    }
  }
}

__global__ __launch_bounds__(256) void quantum_head_kernel(
    const float* __restrict__ x,      // (B, 512)
    const float* __restrict__ fcw,    // (4, 512)
    const float* __restrict__ fcb,    // (4,)
    const float* __restrict__ wrot,   // (2, 4, 3)
    float* __restrict__ out) {        // (B,)
  // B fragments for all 128 K-chunks, in exact WMMA lane layout: 32 KB
  __shared__ v2f   bfrag[KCHUNKS * 32];
  __shared__ float ang[ROWS_PER_BLOCK][4];  // per-row angles after tanh
  __shared__ float wsh[24];                 // Rot parameters

  const int tid  = threadIdx.x;
  const int lane = tid & 31;
  const int wave = tid >> 5;           // 0..7

  if (tid < 24) wsh[tid] = wrot[tid];

  // ---- Stage fc_w (padded 4->16 cols) into LDS as per-chunk B fragments ----
  // B (4x16, KxN) lane layout: lane l holds N=l%16, K = 2*(l/16) + {vgpr0, vgpr1}
#pragma unroll 4
  for (int e = tid; e < KCHUNKS * 32; e += 256) {
    int c    = e >> 5;
    int l    = e & 31;
    int col  = l & 15;
    int k0   = 4 * c + 2 * (l >> 4);
    v2f v;
    if (col < NQ) {
      v.x = fcw[col * DIN + k0];
      v.y = fcw[col * DIN + k0 + 1];
    } else {
      v.x = 0.0f; v.y = 0.0f;
    }
    bfrag[e] = v;
  }
  __syncthreads();

  // ---- Phase 1: skinny GEMM via V_WMMA_F32_16X16X4_F32 ----
  // A (16x4) lane layout: lane l holds M=l%16, K = 2*(l/16) + {vgpr0, vgpr1}
  const int m    = lane & 15;
  const int half = lane >> 4;
  const int rowBase = blockIdx.x * ROWS_PER_BLOCK + wave * 16;
  const float* xrow = x + (size_t)(rowBase + m) * DIN + 2 * half;

  v8f acc = {0.f, 0.f, 0.f, 0.f, 0.f, 0.f, 0.f, 0.f};

  for (int cc = 0; cc < KCHUNKS; cc += 8) {
    if (cc + 8 < KCHUNKS)
      __builtin_prefetch(xrow + 4 * (cc + 8), 0, 3);   // WGP-scope: pull next 128B line into near caches
#pragma unroll
    for (int c = cc; c < cc + 8; ++c) {
      v2f a = *(const v2f*)(xrow + 4 * c);             // 8B-aligned global_load_b64
      v2f b = bfrag[c * 32 + lane];                    // conflict-free ds_load_b64
      acc = __builtin_amdgcn_wmma_f32_16x16x4_f32(
          /*neg_a=*/false, a, /*neg_b=*/false, b,
          /*c_mod=*/(short)0, acc, /*reuse_a=*/false, /*reuse_b=*/false);
    }
  }

  // C/D layout: VGPR v holds M = v + 8*(lane/16), N = lane%16. Only N<4 is real.
  if (m < NQ) {
    float bias = fcb[m];
#pragma unroll
    for (int v = 0; v < 8; ++v) {
      int r = wave * 16 + v + 8 * half;
      ang[r][m] = tanhf(acc[v] + bias);
    }
  }
  __syncthreads();

  // ---- Phase 2: 4-qubit statevector simulation, one thread per row ----
  if (tid < ROWS_PER_BLOCK) {
    const int row = blockIdx.x * ROWS_PER_BLOCK + tid;
    float A[4];
#pragma unroll
    for (int q = 0; q < 4; ++q) A[q] = ang[tid][q];

    float sr[16], si[16];
#pragma unroll
    for (int i = 0; i < 16; ++i) { sr[i] = 0.f; si[i] = 0.f; }
    sr[0] = 1.f;

    // AngleEmbedding: RX(A[q]) on wire q (wire 0 = MSB => bit = 8>>q)
#pragma unroll
    for (int q = 0; q < 4; ++q) {
      float c = __cosf(0.5f * A[q]);
      float s = __sinf(0.5f * A[q]);
      apply_gate(sr, si, 8 >> q,
                 c, 0.f, 0.f, -s,
                 0.f, -s, c, 0.f);
    }

    // StronglyEntanglingLayers: 2 layers of Rot + CNOT ring
#pragma unroll
    for (int l = 0; l < 2; ++l) {
#pragma unroll
      for (int q = 0; q < 4; ++q) {
        float ph = wsh[l * 12 + q * 3 + 0];
        float th = wsh[l * 12 + q * 3 + 1];
        float om = wsh[l * 12 + q * 3 + 2];
        float ct  = __cosf(0.5f * th), st = __sinf(0.5f * th);
        float ap  = 0.5f * (ph + om),  am = 0.5f * (ph - om);
        float cap = __cosf(ap), sap = __sinf(ap);
        float cam = __cosf(am), sam = __sinf(am);
        // U = [[e^{-i ap} ct, -e^{+i am} st], [e^{-i am} st, e^{+i ap} ct]]
        apply_gate(sr, si, 8 >> q,
                   cap * ct, -sap * ct,  -cam * st, -sam * st,
                   cam * st, -sam * st,   cap * ct,  sap * ct);
      }
      const int r = (l % 3) + 1;  // ranges: layer0 -> 1, layer1 -> 2
#pragma unroll
      for (int q = 0; q < 4; ++q)
        cnot_gate(sr, si, 8 >> q, 8 >> ((q + r) & 3));
    }

    // mean_i <Z_i> = sum_idx |amp|^2 * (1 - popcount(idx)/2)
    float acc_z = 0.f;
#pragma unroll
    for (int i = 0; i < 16; ++i) {
      float p = sr[i] * sr[i] + si[i] * si[i];
      acc_z += p * (1.0f - 0.5f * (float)__popc(i));
    }
    out[row] = acc_z;
  }
}

extern "C" void kernel_launch(void* const* d_in, const int* in_sizes, int n_in,
                              void* d_out, int out_size, void* d_ws, size_t ws_size,
                              hipStream_t stream) {
  const float* x   = (const float*)d_in[0];  // (B, 512)
  const float* fcw = (const float*)d_in[1];  // (4, 512)
  const float* fcb = (const float*)d_in[2];  // (4,)
  const float* w   = (const float*)d_in[3];  // (2, 4, 3)
  float* out = (float*)d_out;

  const int rows   = in_sizes[0] / DIN;             // 65536
  const int blocks = rows / ROWS_PER_BLOCK;         // 512

  quantum_head_kernel<<<blocks, 256, 0, stream>>>(x, fcw, fcb, w, out);
}